// OHEMloss_75359496175834
// MI455X (gfx1250) — compile-verified
//
#include <hip/hip_runtime.h>
#include <cstdint>
#include <cstddef>

// ---------------- problem constants (match reference) ----------------
#define B_  32
#define N_  131072
#define L_  10
#define UP_CONF 0.8f
#define LO_CONF 0.3f
#define UP_IOU  0.5f
#define LO_IOU  0.35f
#define NUM_NEG_RATIO 3
#define MAX_OBJS 10

// ---------------- tiling ----------------
#define THREADS      256
#define TILE_ROWS    1024                       // rows per LDS tile
#define F4_PER_TILE  (TILE_ROWS * 5 / 4)        // 1280 float4 = 20480 B
#define PASSES       (F4_PER_TILE / THREADS)    // 5 b128 per thread per tile
#define TILES_PER_BLOCK 4
#define ROWS_PER_BLOCK  (TILE_ROWS * TILES_PER_BLOCK)  // 4096
#define NB  (N_ / ROWS_PER_BLOCK)               // 32 blocks per image

static_assert(NB == 32, "phase-2 merge assumes 32 blocks per image (one wave)");

struct BlockRec {
  float posIou[L_];
  int   posIdx[L_];
  float negIou[L_];
  int   lowCnt;
  int   lowIdx[L_];
  int   anyUpper;
};
struct ImgOut { float conf; float reg; int npos; };

// ---------------- CDNA5 async global->LDS path (guarded) ----------------
#if defined(__has_builtin)
#  if __has_builtin(__builtin_amdgcn_global_load_async_to_lds_b128) && \
      __has_builtin(__builtin_amdgcn_s_wait_asynccnt)
#    define USE_ASYNC 1
#  endif
#endif
#ifndef USE_ASYNC
#  define USE_ASYNC 0
#endif

#if USE_ASYNC
typedef int v4i_ __attribute__((ext_vector_type(4)));
#endif

__device__ __forceinline__ void load_tile(float4* dstLds, const float4* srcGlb, int tid) {
#if USE_ASYNC
  typedef __attribute__((address_space(1))) v4i_ GVec;   // "__device__" int4
  typedef __attribute__((address_space(3))) v4i_ LVec;   // "__shared__" int4
  GVec* gp = (GVec*)(uintptr_t)(const void*)srcGlb;      // flat->global: identical bits
  LVec* lp = (LVec*)(unsigned)(uintptr_t)(void*)dstLds;  // flat LDS low 32 bits == LDS offset
  #pragma unroll
  for (int p = 0; p < PASSES; ++p) {
    __builtin_amdgcn_global_load_async_to_lds_b128(gp + (p * THREADS + tid),
                                                   lp + (p * THREADS + tid),
                                                   0, 0);
  }
#else
  #pragma unroll
  for (int p = 0; p < PASSES; ++p)
    dstLds[p * THREADS + tid] = srcGlb[p * THREADS + tid];   // global_load_b128 + ds_store_b128
#endif
}

__device__ __forceinline__ void wait_tile(bool more_in_flight) {
#if USE_ASYNC
  if (more_in_flight) __builtin_amdgcn_s_wait_asynccnt(PASSES);
  else                __builtin_amdgcn_s_wait_asynccnt(0);
#else
  (void)more_in_flight;
#endif
}

// ---------------- math helpers (match jax.nn semantics) ----------------
__device__ __forceinline__ float softplus_f(float x) {   // logaddexp(x, 0)
  return fmaxf(x, 0.0f) + log1pf(expf(-fabsf(x)));
}
__device__ __forceinline__ float sigmoid_f(float x) {    // stable, branchless
  float e = expf(-fabsf(x));
  return ((x >= 0.0f) ? 1.0f : e) / (1.0f + e);
}
__device__ __forceinline__ float sl1(float d) {
  float a = fabsf(d);
  return a < 1.0f ? 0.5f * d * d : a - 0.5f;
}

// =====================================================================
// Phase 1: stream outputs, per-label masked max/argmax + first-10 lower
// IoU maxes tracked as (inter, denom) pairs; one divide per label at end.
// =====================================================================
__global__ __launch_bounds__(THREADS)
void ohem_phase1(const float* __restrict__ outs, const float* __restrict__ labels,
                 BlockRec* __restrict__ recs)
{
  const int tid = threadIdx.x;
  const int b   = blockIdx.x / NB;
  const int blk = blockIdx.x % NB;
  const int rowBase = blk * ROWS_PER_BLOCK;

  __shared__ float4 sBuf[2][F4_PER_TILE];      // 40960 B double buffer
  __shared__ float  sG[L_][5];                 // label x1,y1,x2,y2,area
  __shared__ int    sLowCnt;
  __shared__ int    sLowIdx[L_];
  __shared__ int    sWaveCnt[8];
  __shared__ float  sBN[8][L_];                // best-pos inter
  __shared__ float  sBD[8][L_];                // best-pos denom
  __shared__ int    sBX[8][L_];                // best-pos idx
  __shared__ float  sNN[8][L_];                // best-neg inter
  __shared__ float  sND[8][L_];                // best-neg denom
  __shared__ int    sRA[8];

  if (tid < L_) {
    const float* lb = labels + ((size_t)b * L_ + tid) * 4;
    float cx = lb[0], cy = lb[1], w = lb[2], h = lb[3];
    sG[tid][0] = cx - w * 0.5f; sG[tid][1] = cy - h * 0.5f;
    sG[tid][2] = cx + w * 0.5f; sG[tid][3] = cy + h * 0.5f;
    sG[tid][4] = w * h;
  }
  if (tid == 0) sLowCnt = 0;

  const float4* src = (const float4*)(outs + ((size_t)b * N_ + rowBase) * 5);
  load_tile(&sBuf[0][0], src, tid);            // prime the pipeline

  __syncthreads();                             // sG visible to everyone
  // label geometry in registers -> zero LDS traffic in the hot loop
  float gx1[L_], gy1[L_], gx2[L_], gy2[L_], gA[L_];
  #pragma unroll
  for (int l = 0; l < L_; ++l) {
    gx1[l] = sG[l][0]; gy1[l] = sG[l][1];
    gx2[l] = sG[l][2]; gy2[l] = sG[l][3]; gA[l] = sG[l][4];
  }

  // max trackers as (numerator=inter, denominator) pairs; sentinel = -1/1
  float bN[L_], bD[L_]; int bX[L_];
  float nN[L_], nD[L_];
  #pragma unroll
  for (int l = 0; l < L_; ++l) {
    bN[l] = -1.0f; bD[l] = 1.0f; bX[l] = 0x7fffffff;
    nN[l] = -1.0f; nD[l] = 1.0f;
  }
  int anyUp = 0;

  for (int t = 0; t < TILES_PER_BLOCK; ++t) {
    if (t + 1 < TILES_PER_BLOCK) {
      load_tile(&sBuf[(t + 1) & 1][0], src + (size_t)(t + 1) * F4_PER_TILE, tid);
      wait_tile(true);
    } else {
      wait_tile(false);
    }
    if (t + 2 < TILES_PER_BLOCK)               // global_prefetch_b8, one line/thread
      __builtin_prefetch((const char*)src + (size_t)(t + 2) * (F4_PER_TILE * 16) + tid * 80, 0, 1);
    __syncthreads();

    const float* bf = (const float*)&sBuf[t & 1][0];

    for (int j = 0; j < TILE_ROWS / THREADS; ++j) {
      const int r   = j * THREADS + tid;
      const int row = rowBase + t * TILE_ROWS + r;
      float pcx = bf[r * 5 + 0], pcy = bf[r * 5 + 1];
      float pw  = bf[r * 5 + 2], ph  = bf[r * 5 + 3];
      float lg  = bf[r * 5 + 4];
      float conf = sigmoid_f(lg);
      bool up = conf > UP_CONF;
      bool lo = conf < LO_CONF;
      anyUp |= (int)up;

      if (up || lo) {
        float px1 = pcx - pw * 0.5f, py1 = pcy - ph * 0.5f;
        float px2 = pcx + pw * 0.5f, py2 = pcy + ph * 0.5f;
        float pa  = pw * ph;
        float inter[L_];
        bool any = false;
        #pragma unroll
        for (int l = 0; l < L_; ++l) {         // pass 1: cheap geometry
          float iw = fminf(px2, gx2[l]) - fmaxf(px1, gx1[l]);
          float ih = fminf(py2, gy2[l]) - fmaxf(py1, gy1[l]);
          float in_ = fmaxf(iw, 0.0f) * fmaxf(ih, 0.0f);  // == iou numerator w/ en-mask
          inter[l] = in_;
          any = any || (in_ > 0.0f);
        }
        if (any) {                             // pass 2: updates; most waves skip
          #pragma unroll
          for (int l = 0; l < L_; ++l) {
            float dn = pa + gA[l] - inter[l];  // > 0 always (pa > 0)
            // iou_new > iou_best  <=>  inter*bD > bN*dn   (both denoms > 0)
            if (up && inter[l] * bD[l] > bN[l] * dn) {
              bN[l] = inter[l]; bD[l] = dn; bX[l] = row;   // strict > keeps first index
            }
            if (lo && inter[l] * nD[l] > nN[l] * dn) {
              nN[l] = inter[l]; nD[l] = dn;
            }
          }
        }
      }

      // ordered collection of the first 10 "lower" indices (block-uniform branch)
      if (sLowCnt < L_) {
        unsigned bal = (unsigned)__ballot(lo);
        int lane = tid & 31, wv = tid >> 5;
        if (lane == 0) sWaveCnt[wv] = __popc(bal);
        __syncthreads();
        int base = sLowCnt;
        int woff = 0, tot = 0;
        #pragma unroll
        for (int w = 0; w < 8; ++w) { int c = sWaveCnt[w]; if (w < wv) woff += c; tot += c; }
        int pos = base + woff + __popc(bal & ((1u << lane) - 1u));
        if (lo && pos < L_) sLowIdx[pos] = row;
        __syncthreads();
        if (tid == 0) sLowCnt = (base + tot > L_) ? L_ : (base + tot);
        __syncthreads();
      }
    }
    __syncthreads();    // everyone done reading this buffer before it is refilled
  }

  // ---- block reduction: wave32 shuffles on (inter, denom, idx) ----
  #pragma unroll
  for (int off = 16; off > 0; off >>= 1) {
    #pragma unroll
    for (int l = 0; l < L_; ++l) {
      float oN = __shfl_down(bN[l], off);
      float oD = __shfl_down(bD[l], off);
      int   ox = __shfl_down(bX[l], off);
      float lhs = oN * bD[l], rhs = bN[l] * oD;
      if (lhs > rhs || (lhs == rhs && ox < bX[l])) { bN[l] = oN; bD[l] = oD; bX[l] = ox; }
      float qN = __shfl_down(nN[l], off);
      float qD = __shfl_down(nD[l], off);
      if (qN * nD[l] > nN[l] * qD) { nN[l] = qN; nD[l] = qD; }
    }
    anyUp |= __shfl_down(anyUp, off);
  }
  {
    int lane = tid & 31, wv = tid >> 5;
    if (lane == 0) {
      #pragma unroll
      for (int l = 0; l < L_; ++l) {
        sBN[wv][l] = bN[l]; sBD[wv][l] = bD[l]; sBX[wv][l] = bX[l];
        sNN[wv][l] = nN[l]; sND[wv][l] = nD[l];
      }
      sRA[wv] = anyUp;
    }
  }
  __syncthreads();

  BlockRec* rec = &recs[blockIdx.x];
  if (tid < L_) {
    float mN = sBN[0][tid], mD = sBD[0][tid]; int mx = sBX[0][tid];
    float qN = sNN[0][tid], qD = sND[0][tid];
    for (int w = 1; w < 8; ++w) {
      float oN = sBN[w][tid], oD = sBD[w][tid]; int ox = sBX[w][tid];
      float lhs = oN * mD, rhs = mN * oD;
      if (lhs > rhs || (lhs == rhs && ox < mx)) { mN = oN; mD = oD; mx = ox; }
      float pN = sNN[w][tid], pD = sND[w][tid];
      if (pN * qD > qN * pD) { qN = pN; qD = pD; }
    }
    rec->posIou[tid] = mN / mD;       // one divide per label per block
    rec->posIdx[tid] = mx;
    rec->negIou[tid] = qN / qD;
  }
  if (tid == 0) {
    int a = 0;
    for (int w = 0; w < 8; ++w) a |= sRA[w];
    rec->anyUpper = a;
    int c = sLowCnt;
    rec->lowCnt = c;
    for (int k = 0; k < c; ++k) rec->lowIdx[k] = sLowIdx[k];
  }
}

// =====================================================================
// Phase 2: merge 32 block records per image + O(L^2) epilogue
// =====================================================================
__global__ __launch_bounds__(32)
void ohem_phase2(const float* __restrict__ outs, const float* __restrict__ labels,
                 const BlockRec* __restrict__ recs, ImgOut* __restrict__ img)
{
  const int b    = blockIdx.x;
  const int lane = threadIdx.x;        // 32 lanes == NB
  const BlockRec* r = &recs[b * NB + lane];

  float pI[L_]; int pX[L_]; float nI[L_];
  #pragma unroll
  for (int l = 0; l < L_; ++l) { pI[l] = r->posIou[l]; pX[l] = r->posIdx[l]; nI[l] = r->negIou[l]; }
  int aUp = r->anyUpper;

  #pragma unroll
  for (int off = 16; off > 0; off >>= 1) {
    #pragma unroll
    for (int l = 0; l < L_; ++l) {
      float oi = __shfl_down(pI[l], off);
      int   ox = __shfl_down(pX[l], off);
      if (oi > pI[l] || (oi == pI[l] && ox < pX[l])) { pI[l] = oi; pX[l] = ox; }
      nI[l] = fmaxf(nI[l], __shfl_down(nI[l], off));
    }
    aUp |= __shfl_down(aUp, off);
  }
  if (lane != 0) return;

  // merge ordered "first lower" lists in block order
  int fl[L_];
  #pragma unroll
  for (int k = 0; k < L_; ++k) fl[k] = 0;    // argmax-of-all-false fallback = 0
  int tot = 0, anyLow = 0;
  for (int blk = 0; blk < NB && tot < L_; ++blk) {
    const BlockRec* rb = &recs[b * NB + blk];
    int c = rb->lowCnt;
    if (c > 0) anyLow = 1;
    for (int k = 0; k < c && tot < L_; ++k) fl[tot++] = rb->lowIdx[k];
  }

  // labels
  float gx[L_], gy[L_], gw[L_], gh[L_];
  #pragma unroll
  for (int l = 0; l < L_; ++l) {
    const float* lb = labels + ((size_t)b * L_ + l) * 4;
    gx[l] = lb[0]; gy[l] = lb[1]; gw[l] = lb[2]; gh[l] = lb[3];
  }
  int last = L_;
  for (int l = 0; l < L_; ++l) if (gh[l] == 0.0f) { last = l; break; }

  const float* ob = outs + (size_t)b * N_ * 5;
  float confPos = 0.0f, reg = 0.0f;
  int numPos = 0;
  bool  ncand[L_]; float cand[L_], lgF[L_];

  for (int l = 0; l < L_; ++l) {
    bool gtv = (l < last);
    if (gtv && pI[l] >= UP_IOU) {
      ++numPos;
      const float* o = ob + (size_t)pX[l] * 5;
      confPos += softplus_f(-o[4]);
      float s = sl1(o[0] - gx[l]) + sl1(o[1] - gy[l]) + sl1(o[2] - gw[l]) + sl1(o[3] - gh[l]);
      float ml = fmaxf(gw[l], gh[l]); if (ml == 0.0f) ml = 1.0f;
      reg += s / ml;
    }
    bool nc = gtv && (nI[l] <= LO_IOU);
    ncand[l] = nc;
    float lg = ob[(size_t)fl[l] * 5 + 4];
    lgF[l] = lg;
    cand[l] = nc ? sigmoid_f(lg) : -__builtin_inff();
  }

  float confNeg = 0.0f;
  const int lim = NUM_NEG_RATIO * numPos;
  for (int l = 0; l < L_; ++l) {
    if (!ncand[l]) continue;
    int rank = 0;      // stable descending rank == argsort(argsort(-cand))
    for (int j = 0; j < L_; ++j)
      if (cand[j] > cand[l] || (cand[j] == cand[l] && j < l)) ++rank;
    if (rank < lim) confNeg += softplus_f(lgF[l]);
  }

  bool valid = (last <= MAX_OBJS) && (aUp != 0) && (anyLow != 0) && (numPos > 0);
  img[b].conf = valid ? (confPos + confNeg) : 0.0f;
  img[b].reg  = valid ? reg : 0.0f;
  img[b].npos = valid ? numPos : 0;
}

// =====================================================================
// Phase 3: reduce 32 images, write [total, conf, reg, pos]
// =====================================================================
__global__ __launch_bounds__(32)
void ohem_phase3(const ImgOut* __restrict__ img, float* __restrict__ out)
{
  int lane = threadIdx.x;
  float c = img[lane].conf;
  float r = img[lane].reg;
  int   p = img[lane].npos;
  #pragma unroll
  for (int off = 16; off > 0; off >>= 1) {
    c += __shfl_down(c, off);
    r += __shfl_down(r, off);
    p += __shfl_down(p, off);
  }
  if (lane == 0) {
    bool ok = p >= 1;
    float pf = (float)(p > 1 ? p : 1);
    out[0] = ok ? (c + 1.0f * r) / pf : 0.0f;
    out[1] = ok ? c / pf : 0.0f;
    out[2] = ok ? r / pf : 0.0f;
    out[3] = ok ? (float)p : 0.0f;
  }
}

// =====================================================================
extern "C" void kernel_launch(void* const* d_in, const int* in_sizes, int n_in,
                              void* d_out, int out_size, void* d_ws, size_t ws_size,
                              hipStream_t stream) {
  const float* labels  = (const float*)d_in[0];   // (B, L, 4)
  const float* outputs = (const float*)d_in[1];   // (B, N, 5)
  float* out = (float*)d_out;                     // 4 floats

  BlockRec* recs = (BlockRec*)d_ws;
  ImgOut*   img  = (ImgOut*)((char*)d_ws + sizeof(BlockRec) * (size_t)B_ * NB);

  ohem_phase1<<<dim3(B_ * NB), dim3(THREADS), 0, stream>>>(outputs, labels, recs);
  ohem_phase2<<<dim3(B_), dim3(32), 0, stream>>>(outputs, labels, recs, img);
  ohem_phase3<<<dim3(1), dim3(32), 0, stream>>>(img, out);
}